// Triplet_Loss_86311662780622
// MI455X (gfx1250) — compile-verified
//
#include <hip/hip_runtime.h>
#include <hip/hip_bf16.h>
#include <math.h>

// ---------------------------------------------------------------------------
// Triplet loss, fused:  cross = img @ txt^T  (f32 WMMA 16x16x4),
// squared pairwise distance epilogue, masked row-min + diagonal, final reduce.
// B = 8192 rows, D = 512 features, all float32.
// GEMM: 32x64 tile per wave (2x4 WMMA accumulators), 2-stage register
// pipeline, single base pointer per matrix with immediate offsets so all
// fragment loads lower to global_load_b64 with 24-bit IOFFSET.
// ---------------------------------------------------------------------------

#define NB   8192
#define ND   512
#define TL_EPS    1e-6f
#define TL_MARGIN 1.0f

typedef __attribute__((ext_vector_type(2))) float v2f;
typedef __attribute__((ext_vector_type(8))) float v8f;

// ---------------------------------------------------------------------------
// Kernel 1: per-row sums / sum-of-squares for img and txt rows.
//   rowterm_a[i] = ||img_i||^2 + 2*eps*sum(img_i)
//   colterm_t[j] = ||txt_j||^2 - 2*eps*sum(txt_j)
// Also initializes rowmin_bits[i] = +inf (bit pattern) for the atomic min.
// ---------------------------------------------------------------------------
__global__ void __launch_bounds__(256)
tl_row_stats(const float* __restrict__ img, const float* __restrict__ txt,
             float* __restrict__ rowterm_a, float* __restrict__ colterm_t,
             unsigned int* __restrict__ rowmin_bits)
{
    const int row = blockIdx.x;        // 0..NB-1
    const int tid = threadIdx.x;       // 0..255
    const float* ai = img + (size_t)row * ND;
    const float* ti = txt + (size_t)row * ND;

    float s_a = 0.f, q_a = 0.f, s_t = 0.f, q_t = 0.f;
    for (int c = tid; c < ND; c += 256) {
        float va = ai[c], vt = ti[c];
        s_a += va; q_a += va * va;
        s_t += vt; q_t += vt * vt;
    }
    for (int off = 16; off > 0; off >>= 1) {
        s_a += __shfl_down(s_a, off, 32);
        q_a += __shfl_down(q_a, off, 32);
        s_t += __shfl_down(s_t, off, 32);
        q_t += __shfl_down(q_t, off, 32);
    }
    __shared__ float red[4][8];
    const int wid = tid >> 5, lane = tid & 31;
    if (lane == 0) {
        red[0][wid] = s_a; red[1][wid] = q_a;
        red[2][wid] = s_t; red[3][wid] = q_t;
    }
    __syncthreads();
    if (tid == 0) {
        float SA = 0.f, QA = 0.f, ST = 0.f, QT = 0.f;
        #pragma unroll
        for (int w = 0; w < 8; ++w) {
            SA += red[0][w]; QA += red[1][w];
            ST += red[2][w]; QT += red[3][w];
        }
        rowterm_a[row]   = QA + 2.0f * TL_EPS * SA;
        colterm_t[row]   = QT - 2.0f * TL_EPS * ST;
        rowmin_bits[row] = 0x7F800000u;   // +inf
    }
}

// ---------------------------------------------------------------------------
// Kernel 2: fused GEMM + distance epilogue.
// One wave computes a 32(M) x 64(N) tile of cross = img @ txt^T via 2x4
// V_WMMA_F32_16X16X4_F32 accumulators; K is software-pipelined (2 stages).
// All fragment loads are base + compile-time offset (sub-tile row stride
// 16*ND*4 = 32 KB fits the 24-bit signed IOFFSET).
// ---------------------------------------------------------------------------
__global__ void __launch_bounds__(256)
tl_gemm_min(const float* __restrict__ img, const float* __restrict__ txt,
            const float* __restrict__ rowterm_a, const float* __restrict__ colterm_t,
            unsigned int* __restrict__ rowmin_bits, float* __restrict__ diag_sq)
{
    const int lane = threadIdx.x & 31;
    const int wid  = blockIdx.x * (blockDim.x >> 5) + (threadIdx.x >> 5);
    // (NB/32) x (NB/64) = 256 x 128 = 32768 waves total
    const int iT = wid & 255;
    const int jT = wid >> 8;
    const int I0 = iT * 32;
    const int J0 = jT * 64;

    const int m = lane & 15;            // M index for A rows, N index for B / C-D
    const int g = lane >> 4;            // half-wave selector (K pair / M+8)

    // Single base pointer per matrix; sub-tile rows addressed by immediates.
    const float* aB = img + (size_t)(I0 + m) * ND + 2 * g;
    const float* bB = txt + (size_t)(J0 + m) * ND + 2 * g;

    v8f acc[2][4] = {};                 // 8 accumulators (64 VGPRs)
    v2f a0[2], b0[4], a1[2], b1[4];     // 2-stage fragment buffers (24 VGPRs)

    // Preload stage 0 (k = 0)
    #pragma unroll
    for (int mt = 0; mt < 2; ++mt) a0[mt] = *(const v2f*)(aB + mt * 16 * ND);
    #pragma unroll
    for (int nt = 0; nt < 4; ++nt) b0[nt] = *(const v2f*)(bB + nt * 16 * ND);

    // Main pipeline: kb = 0, 8, ..., 496  (stage0 = kb, stage1 = kb+4)
    for (int kb = 0; kb < ND - 8; kb += 8) {
        // Issue stage-1 loads (k = kb+4)
        #pragma unroll
        for (int mt = 0; mt < 2; ++mt) a1[mt] = *(const v2f*)(aB + mt * 16 * ND + 4);
        #pragma unroll
        for (int nt = 0; nt < 4; ++nt) b1[nt] = *(const v2f*)(bB + nt * 16 * ND + 4);
        // 8 WMMAs on stage 0
        #pragma unroll
        for (int mt = 0; mt < 2; ++mt)
            #pragma unroll
            for (int nt = 0; nt < 4; ++nt)
                acc[mt][nt] = __builtin_amdgcn_wmma_f32_16x16x4_f32(
                    false, a0[mt], false, b0[nt], (short)0, acc[mt][nt], false, false);
        // Issue next stage-0 loads (k = kb+8)
        #pragma unroll
        for (int mt = 0; mt < 2; ++mt) a0[mt] = *(const v2f*)(aB + mt * 16 * ND + 8);
        #pragma unroll
        for (int nt = 0; nt < 4; ++nt) b0[nt] = *(const v2f*)(bB + nt * 16 * ND + 8);
        // 8 WMMAs on stage 1
        #pragma unroll
        for (int mt = 0; mt < 2; ++mt)
            #pragma unroll
            for (int nt = 0; nt < 4; ++nt)
                acc[mt][nt] = __builtin_amdgcn_wmma_f32_16x16x4_f32(
                    false, a1[mt], false, b1[nt], (short)0, acc[mt][nt], false, false);
        // Advance bases by 8 columns (32 bytes)
        aB += 8;
        bB += 8;
    }

    // Tail: stage0 holds k = 504; load k = 508 (no out-of-bounds prefetch)
    #pragma unroll
    for (int mt = 0; mt < 2; ++mt) a1[mt] = *(const v2f*)(aB + mt * 16 * ND + 4);
    #pragma unroll
    for (int nt = 0; nt < 4; ++nt) b1[nt] = *(const v2f*)(bB + nt * 16 * ND + 4);
    #pragma unroll
    for (int mt = 0; mt < 2; ++mt)
        #pragma unroll
        for (int nt = 0; nt < 4; ++nt)
            acc[mt][nt] = __builtin_amdgcn_wmma_f32_16x16x4_f32(
                false, a0[mt], false, b0[nt], (short)0, acc[mt][nt], false, false);
    #pragma unroll
    for (int mt = 0; mt < 2; ++mt)
        #pragma unroll
        for (int nt = 0; nt < 4; ++nt)
            acc[mt][nt] = __builtin_amdgcn_wmma_f32_16x16x4_f32(
                false, a1[mt], false, b1[nt], (short)0, acc[mt][nt], false, false);

    // ------------------------------------------------------------------
    // Epilogue.  C/D layout: element (M = 16*mt + r + 8*g, N = 16*nt + m)
    // lives in acc[mt][nt][r].
    // ------------------------------------------------------------------
    const float Ceps = (float)ND * TL_EPS * TL_EPS;
    float ct[4];
    #pragma unroll
    for (int nt = 0; nt < 4; ++nt) ct[nt] = colterm_t[J0 + 16 * nt + m];

    #pragma unroll
    for (int mt = 0; mt < 2; ++mt) {
        #pragma unroll
        for (int r = 0; r < 8; ++r) {
            const int Mrow = I0 + 16 * mt + r + 8 * g;
            const float rt = rowterm_a[Mrow];
            float best = __builtin_huge_valf();
            #pragma unroll
            for (int nt = 0; nt < 4; ++nt) {
                const int col = J0 + 16 * nt + m;
                float sq = rt + ct[nt] - 2.0f * acc[mt][nt][r] + Ceps;
                sq = fmaxf(sq, 0.0f);
                if (col == Mrow) {
                    diag_sq[Mrow] = sq;            // exactly one writer per row
                } else {
                    best = fminf(best, sq);
                }
            }
            // min across the 16 lanes sharing this row (xor < 16 stays in half)
            #pragma unroll
            for (int off = 1; off < 16; off <<= 1)
                best = fminf(best, __shfl_xor(best, off, 32));
            if (m == 0)
                atomicMin(&rowmin_bits[Mrow], __float_as_uint(best));
        }
    }
}

// ---------------------------------------------------------------------------
// Kernel 3: loss = mean(relu(sqrt(diag_sq) - sqrt(rowmin) + margin))
// ---------------------------------------------------------------------------
__global__ void __launch_bounds__(1024)
tl_final_loss(const float* __restrict__ diag_sq,
              const unsigned int* __restrict__ rowmin_bits,
              float* __restrict__ out)
{
    const int tid = threadIdx.x;
    float sum = 0.f;
    for (int i = tid; i < NB; i += 1024) {
        float dpos = sqrtf(diag_sq[i]);
        float dneg = sqrtf(__uint_as_float(rowmin_bits[i]));
        sum += fmaxf(dpos - dneg + TL_MARGIN, 0.0f);
    }
    for (int off = 16; off > 0; off >>= 1)
        sum += __shfl_down(sum, off, 32);
    __shared__ float red[32];
    const int wid = tid >> 5, lane = tid & 31;
    if (lane == 0) red[wid] = sum;
    __syncthreads();
    if (tid == 0) {
        float total = 0.f;
        #pragma unroll
        for (int w = 0; w < 32; ++w) total += red[w];
        out[0] = total / (float)NB;
    }
}

// ---------------------------------------------------------------------------
// Entry point
// ---------------------------------------------------------------------------
extern "C" void kernel_launch(void* const* d_in, const int* in_sizes, int n_in,
                              void* d_out, int out_size, void* d_ws, size_t ws_size,
                              hipStream_t stream) {
    (void)in_sizes; (void)n_in; (void)out_size; (void)ws_size;

    const float* img = (const float*)d_in[0];   // [NB, ND] f32
    const float* txt = (const float*)d_in[1];   // [NB, ND] f32
    // d_in[2] = logit_scale, d_in[3] = labels: unused by the loss.
    float* out = (float*)d_out;                 // scalar f32

    // Workspace layout (128 KB total)
    float*        rowterm_a   = (float*)d_ws;                 // NB floats
    float*        colterm_t   = rowterm_a + NB;               // NB floats
    unsigned int* rowmin_bits = (unsigned int*)(colterm_t + NB); // NB u32
    float*        diag_sq     = (float*)(rowmin_bits + NB);   // NB floats

    // 1) Row statistics + rowmin init (re-initialized every call: deterministic)
    tl_row_stats<<<NB, 256, 0, stream>>>(img, txt, rowterm_a, colterm_t, rowmin_bits);

    // 2) Fused WMMA GEMM + masked row-min: 32768 waves = 4096 blocks x 8 waves
    tl_gemm_min<<<(NB / 32) * (NB / 64) / 8, 256, 0, stream>>>(
        img, txt, rowterm_a, colterm_t, rowmin_bits, diag_sq);

    // 3) Final scalar reduction
    tl_final_loss<<<1, 1024, 0, stream>>>(diag_sq, rowmin_bits, out);
}